// SpectralCoherence_17368847745225
// MI455X (gfx1250) — compile-verified
//
#include <hip/hip_runtime.h>
#include <math.h>

typedef float v2f __attribute__((ext_vector_type(2)));
typedef float v8f __attribute__((ext_vector_type(8)));

#define NG    201          // grid points
#define NGP   208          // padded to 13*16
#define ND    64           // Delta_y / Delta_z length
#define NCHUNK 13          // NGP / 16

// workspace layout (float offsets)
#define WS_G   0
#define WS_W   (NGP)
#define WS_C2  (2*NGP)                    // C2[i*64+y] = cos(g_i * Dy_y)
#define WS_S2  (WS_C2 + NGP*ND)           // S2[i*64+y] = sin(g_i * Dy_y)
#define WS_B3  (WS_S2 + NGP*ND)           // B3[j*128 + z] = cos(g_j*Dz_z); [j*128+64+z] = sin
#define WS_TOTAL (WS_B3 + NGP*2*ND)

#define LOG2_10 3.32192809488736235f

// ---------------------------------------------------------------------------
// Precompute grid, trapezoid weights, and trig tables (tiny; 1 block).
// ---------------------------------------------------------------------------
__global__ __launch_bounds__(256) void sc_precompute(
    const float* __restrict__ dy, const float* __restrict__ dz,
    float* __restrict__ ws)
{
    const int tid = threadIdx.x;
    float* g = ws + WS_G;
    float* w = ws + WS_W;

    // grid = [-gp[::-1], 0, gp], gp[t] = (10^(-2 + 4t/99))^2
    for (int i = tid; i < NGP; i += 256) {
        float v = 0.f;
        if (i < 100) {
            float e = -2.f + 4.f * (float)(99 - i) / 99.f;
            v = -exp2f(2.f * e * LOG2_10);
        } else if (i == 100) {
            v = 0.f;
        } else if (i <= 200) {
            float e = -2.f + 4.f * (float)(i - 101) / 99.f;
            v = exp2f(2.f * e * LOG2_10);
        }
        g[i] = v;
    }
    __syncthreads();

    // trapezoid weights on the 201-point grid (pad rows -> 0)
    for (int i = tid; i < NGP; i += 256) {
        float v = 0.f;
        if (i < NG) {
            float left  = (i > 0)      ? 0.5f * (g[i] - g[i-1]) : 0.f;
            float right = (i < NG - 1) ? 0.5f * (g[i+1] - g[i]) : 0.f;
            v = left + right;
        }
        w[i] = v;
    }

    // trig tables (g already visible via the earlier barrier)
    for (int idx = tid; idx < NGP * ND; idx += 256) {
        const int i = idx >> 6;
        const int y = idx & 63;
        float c2 = 0.f, s2 = 0.f, c3 = 0.f, s3 = 0.f;
        if (i < NG) {
            const float gv = g[i];
            const float ay = gv * dy[y];
            const float az = gv * dz[y];
            c2 = cosf(ay); s2 = sinf(ay);
            c3 = cosf(az); s3 = sinf(az);
        }
        ws[WS_C2 + idx] = c2;
        ws[WS_S2 + idx] = s2;
        ws[WS_B3 + i*128 + y]      = c3;
        ws[WS_B3 + i*128 + 64 + y] = s3;
    }
}

// ---------------------------------------------------------------------------
// Fused main kernel: one workgroup (8 waves) per k1 sample `a`.
//   chunked over 16 rows of i:
//     phase1: Pw[16][208] -> LDS (transcendental heavy, fp32)
//     phase2: U = Pw * [cos|sin](k3 Dz)   (f32 WMMA 16x16x4, K=208, N=128)
//     phase3: out += C2^T U_r - S2^T U_i  (f32 WMMA, persistent 64x64 accum)
//   F1 reduction folded in; final scale by 1/|F1|.
// ---------------------------------------------------------------------------
__global__ __launch_bounds__(256) void sc_main(
    const float* __restrict__ k1in,
    const float* __restrict__ logL,
    const float* __restrict__ logT,
    const float* __restrict__ logM,
    const float* __restrict__ ws,
    float* __restrict__ out)
{
    __shared__ float lPw[16 * 209];   // stride 209 (odd) -> conflict-free strided A reads
    __shared__ float lU [16 * 132];
    __shared__ float red[256];

    const int a    = blockIdx.x;
    const int tid  = threadIdx.x;
    const int wv   = tid >> 5;       // wave 0..7
    const int lane = tid & 31;
    const int lm   = lane & 15;
    const int hi   = lane >> 4;      // 0: K={0,1}, 1: K={2,3} half of the fragment

    const float k1   = k1in[a];
    const float L    = expf(logL[0]);
    const float T    = expf(logT[0]);
    const float M    = expf(logM[0]);
    const float k1sq = k1 * k1;
    const float L2   = L * L;

    const float* g   = ws + WS_G;
    const float* wt  = ws + WS_W;
    const float* C2m = ws + WS_C2;
    const float* S2m = ws + WS_S2;
    const float* B3  = ws + WS_B3;

    v8f outAcc[2];
    outAcc[0] = (v8f){0.f,0.f,0.f,0.f,0.f,0.f,0.f,0.f};
    outAcc[1] = (v8f){0.f,0.f,0.f,0.f,0.f,0.f,0.f,0.f};
    float f1part = 0.f;

    for (int chunk = 0; chunk < NCHUNK; ++chunk) {
        const int i0 = chunk * 16;

        // ---- phase 1: Pw chunk -> LDS (exactly 13 iters/thread, no divergence)
        for (int idx = tid; idx < 16 * NGP; idx += 256) {
            const int r = idx / NGP;
            const int j = idx - r * NGP;
            const int i = i0 + r;
            float val = 0.f;
            if (i < NG && j < NG) {
                const float k2  = g[i];
                const float k3  = g[j];
                const float kk  = k1sq + k2*k2 + k3*k3;
                const float beta = T * exp2f((-2.f/3.f) * log2f(L * sqrtf(kk)));
                const float k30 = k3 + beta * k1;
                const float kk0 = k1sq + k2*k2 + k30*k30;
                const float k0L2 = L2 * kk0;                       // (L*sqrt(kk0))^2
                const float E0 = M * k0L2 * k0L2 *
                                 exp2f((-17.f/6.f) * log2f(1.f + k0L2));
                const float s   = k1sq + k2*k2;
                const float C1  = beta * k1sq *
                                  (kk0 - 2.f*k30*k30 + beta*k1*k30) / (kk * s);
                const float rs  = sqrtf(s);
                const float C2t = k2 * kk0 / (s * rs) *
                                  atan2f(beta * k1 * rs, kk0 - k30 * k1 * beta);
                const float zeta1 = C1 - (k2 / k1) * C2t;
                const float Phi = E0 / (4.f * 3.14159265358979323f * kk0 * kk0) *
                                  (kk0 - k1sq - 2.f*k1*k30*zeta1 + s*zeta1*zeta1);
                val = Phi * wt[i] * wt[j];
            }
            lPw[r * 209 + j] = val;
            f1part += val;
        }
        __syncthreads();

        // ---- phase 2: U N-tile per wave, 52 fp32 WMMA k-steps
        v8f uacc = (v8f){0.f,0.f,0.f,0.f,0.f,0.f,0.f,0.f};
        const int ncol = wv * 16 + lm;          // 0..127 (cos cols then sin cols)
        #pragma unroll 4
        for (int kk = 0; kk < 52; ++kk) {
            const int kb = kk * 4 + hi * 2;
            v2f af; af.x = lPw[lm * 209 + kb];
                    af.y = lPw[lm * 209 + kb + 1];
            v2f bf; bf.x = B3[kb * 128 + ncol];
                    bf.y = B3[(kb + 1) * 128 + ncol];
            uacc = __builtin_amdgcn_wmma_f32_16x16x4_f32(
                       false, af, false, bf, (short)0, uacc, false, false);
        }
        // scatter U tile to LDS (previous chunk's phase-3 readers already synced)
        #pragma unroll
        for (int r = 0; r < 8; ++r)
            lU[(r + hi * 8) * 132 + ncol] = uacc[r];
        __syncthreads();

        // ---- phase 3: out += C2^T U_r - S2^T U_i  (2 tiles/wave, 4 k-steps)
        #pragma unroll
        for (int t = 0; t < 2; ++t) {
            const int tile = wv + t * 8;
            const int ty = tile >> 2;
            const int tz = tile & 3;
            const int ycol = ty * 16 + lm;
            const int zcol = tz * 16 + lm;
            #pragma unroll
            for (int kk = 0; kk < 4; ++kk) {
                const int kb   = kk * 4 + hi * 2;   // local k row inside chunk
                const int irow = i0 + kb;
                v2f ac; ac.x =  C2m[irow * 64 + ycol];
                        ac.y =  C2m[(irow + 1) * 64 + ycol];
                v2f as; as.x = -S2m[irow * 64 + ycol];     // f32 WMMA has no A-neg
                        as.y = -S2m[(irow + 1) * 64 + ycol];
                v2f br; br.x = lU[kb * 132 + zcol];
                        br.y = lU[(kb + 1) * 132 + zcol];
                v2f bi; bi.x = lU[kb * 132 + 64 + zcol];
                        bi.y = lU[(kb + 1) * 132 + 64 + zcol];
                outAcc[t] = __builtin_amdgcn_wmma_f32_16x16x4_f32(
                                false, ac, false, br, (short)0, outAcc[t], false, false);
                outAcc[t] = __builtin_amdgcn_wmma_f32_16x16x4_f32(
                                false, as, false, bi, (short)0, outAcc[t], false, false);
            }
        }
        __syncthreads();
    }

    // ---- F1 reduction -> den = |F1|
    red[tid] = f1part;
    __syncthreads();
    for (int s = 128; s > 0; s >>= 1) {
        if (tid < s) red[tid] += red[tid + s];
        __syncthreads();
    }
    const float inv = 1.0f / fabsf(red[0]);

    // ---- write 64x64 output tile for this `a`
    #pragma unroll
    for (int t = 0; t < 2; ++t) {
        const int tile = wv + t * 8;
        const int ty = tile >> 2;
        const int tz = tile & 3;
        #pragma unroll
        for (int r = 0; r < 8; ++r) {
            const int y = ty * 16 + r + hi * 8;
            const int z = tz * 16 + lm;
            out[(size_t)a * (ND * ND) + y * ND + z] = outAcc[t][r] * inv;
        }
    }
}

// ---------------------------------------------------------------------------
extern "C" void kernel_launch(void* const* d_in, const int* in_sizes, int n_in,
                              void* d_out, int out_size, void* d_ws, size_t ws_size,
                              hipStream_t stream) {
    const float* k1   = (const float*)d_in[0];   // 512
    const float* dy   = (const float*)d_in[1];   // 64
    const float* dz   = (const float*)d_in[2];   // 64
    const float* logL = (const float*)d_in[3];   // 1
    const float* logT = (const float*)d_in[4];   // 1
    const float* logM = (const float*)d_in[5];   // 1
    float* ws  = (float*)d_ws;
    float* out = (float*)d_out;

    const int NA = in_sizes[0];                  // 512

    sc_precompute<<<1, 256, 0, stream>>>(dy, dz, ws);
    sc_main<<<NA, 256, 0, stream>>>(k1, logL, logT, logM, ws, out);
}